// GCNBlock_20667382628955
// MI455X (gfx1250) — compile-verified
//
#include <hip/hip_runtime.h>

#define NNODES 65536
#define NEDGES 1048576
#define DIM    64

typedef float v2f __attribute__((ext_vector_type(2)));
typedef float v8f __attribute__((ext_vector_type(8)));

// ---------------------------------------------------------------- utilities
__global__ void zero_kernel(float* __restrict__ p, int n) {
    int i = blockIdx.x * blockDim.x + threadIdx.x;
    if (i < n) p[i] = 0.0f;
}

__global__ void degree_kernel(const int* __restrict__ dst, float* __restrict__ deg, int nE) {
    int i = blockIdx.x * blockDim.x + threadIdx.x;
    if (i < nE) atomicAdd(&deg[dst[i]], 1.0f);
}

__global__ void norm_kernel(float* __restrict__ deg, int n) {
    int i = blockIdx.x * blockDim.x + threadIdx.x;
    if (i < n) deg[i] = rsqrtf(fmaxf(deg[i], 1.0f));
}

// o[v][:] = (pw==2 ? norm[v]^2 : norm[v]) * in[v][:]   (float4 per thread)
__global__ void scale_kernel(const float* __restrict__ in, const float* __restrict__ norm,
                             float* __restrict__ o, int nNodes, int pw) {
    int tid = blockIdx.x * blockDim.x + threadIdx.x;
    if (tid >= nNodes * 16) return;
    int v = tid >> 4, c = tid & 15;
    float s = norm[v];
    if (pw == 2) s *= s;
    float4 x = ((const float4*)(in + v * DIM))[c];
    x.x *= s; x.y *= s; x.z *= s; x.w *= s;
    ((float4*)(o + v * DIM))[c] = x;
}

// per edge: AGG[dst] += S[src]  (16 lanes per edge, float4 gather + 4 f32 atomics)
__global__ __launch_bounds__(256) void scatter_kernel(const float* __restrict__ S,
                                                      const int* __restrict__ src,
                                                      const int* __restrict__ dst,
                                                      float* __restrict__ AGG, int nE) {
    int tid = blockIdx.x * blockDim.x + threadIdx.x;
    int e = tid >> 4;
    int c = tid & 15;
    if (e >= nE) return;
    int s = src[e], d = dst[e];
    float4 v = ((const float4*)(S + s * DIM))[c];
    float* p = AGG + d * DIM + c * 4;
    atomicAdd(p + 0, v.x);
    atomicAdd(p + 1, v.y);
    atomicAdd(p + 2, v.z);
    atomicAdd(p + 3, v.w);
}

// ------------------------------------------------- LN + FFN with f32 WMMA
// 8 waves/block, each wave owns a 16-row tile (block covers 128 rows).
__global__ __launch_bounds__(256) void ffn_wmma_kernel(
    const float* __restrict__ r, const float* __restrict__ feat,
    const float* __restrict__ w1, const float* __restrict__ b1,
    const float* __restrict__ w2, const float* __restrict__ b2,
    const float* __restrict__ gamma, const float* __restrict__ beta,
    float* __restrict__ out) {
    __shared__ float lds[8 * 16 * DIM];
    const int wave = threadIdx.x >> 5;
    const int lane = threadIdx.x & 31;
    const int idx  = lane & 15;    // M (A) / N (B,C) index
    const int half = lane >> 4;    // selects K pair (A,B) / +8 row (C)
    const int rowbase = blockIdx.x * 128 + wave * 16;
    float* A = lds + wave * 16 * DIM;

    // ---- LayerNorm of 16 rows into LDS (wave32 shuffle reductions) ----
    const float2 g2  = ((const float2*)gamma)[lane];
    const float2 be2 = ((const float2*)beta)[lane];
    for (int m = 0; m < 16; ++m) {
        float2 x = ((const float2*)(r + (rowbase + m) * DIM))[lane];
        float s = x.x + x.y;
        #pragma unroll
        for (int o = 16; o > 0; o >>= 1) s += __shfl_xor(s, o, 32);
        float mu = s * (1.0f / 64.0f);
        float dx = x.x - mu, dy = x.y - mu;
        float v = dx * dx + dy * dy;
        #pragma unroll
        for (int o = 16; o > 0; o >>= 1) v += __shfl_xor(v, o, 32);
        float rstd = rsqrtf(v * (1.0f / 64.0f) + 1e-5f);
        float2 y;
        y.x = dx * rstd * g2.x + be2.x;
        y.y = dy * rstd * g2.y + be2.y;
        ((float2*)(A + m * DIM))[lane] = y;
    }
    __syncthreads();

    // ---- GEMM1: H = relu(LN @ w1^T + b1) ----
    v8f acc[4];
    #pragma unroll
    for (int j = 0; j < 4; ++j) acc[j] = (v8f)(0.0f);
    for (int kb = 0; kb < 16; ++kb) {
        const int k = kb * 4 + half * 2;
        v2f a = *(const v2f*)(A + idx * DIM + k);            // A[m=idx][k,k+1]
        #pragma unroll
        for (int j = 0; j < 4; ++j) {
            float2 wv = *(const float2*)(w1 + (j * 16 + idx) * DIM + k); // B[k][n]=w1[n][k]
            v2f b = {wv.x, wv.y};
            acc[j] = __builtin_amdgcn_wmma_f32_16x16x4_f32(
                false, a, false, b, (short)0, acc[j], false, false);
        }
    }
    __syncthreads();
    // bias + ReLU, stage H tile back through LDS (re-fragment for GEMM2)
    #pragma unroll
    for (int j = 0; j < 4; ++j) {
        const float bias = b1[j * 16 + idx];
        #pragma unroll
        for (int t = 0; t < 8; ++t) {
            float hv = acc[j][t] + bias;                     // C: row t+8*half, col j*16+idx
            A[(t + 8 * half) * DIM + j * 16 + idx] = fmaxf(hv, 0.0f);
        }
    }
    __syncthreads();

    // ---- GEMM2: FF = H @ w2^T + b2 + r + feat ----
    v8f acc2[4];
    #pragma unroll
    for (int j = 0; j < 4; ++j) acc2[j] = (v8f)(0.0f);
    for (int kb = 0; kb < 16; ++kb) {
        const int k = kb * 4 + half * 2;
        v2f a = *(const v2f*)(A + idx * DIM + k);
        #pragma unroll
        for (int j = 0; j < 4; ++j) {
            float2 wv = *(const float2*)(w2 + (j * 16 + idx) * DIM + k);
            v2f b = {wv.x, wv.y};
            acc2[j] = __builtin_amdgcn_wmma_f32_16x16x4_f32(
                false, a, false, b, (short)0, acc2[j], false, false);
        }
    }
    #pragma unroll
    for (int j = 0; j < 4; ++j) {
        const int col = j * 16 + idx;
        const float bias = b2[col];
        #pragma unroll
        for (int t = 0; t < 8; ++t) {
            const int row = rowbase + t + 8 * half;
            out[row * DIM + col] = acc2[j][t] + bias + r[row * DIM + col] + feat[row * DIM + col];
        }
    }
}

// ---------------------------------------------------------------- launcher
extern "C" void kernel_launch(void* const* d_in, const int* in_sizes, int n_in,
                              void* d_out, int out_size, void* d_ws, size_t ws_size,
                              hipStream_t stream) {
    const float* feat = (const float*)d_in[0];
    const int*   esrc = (const int*)d_in[1];
    const int*   edst = (const int*)d_in[2];
    const float* w1   = (const float*)d_in[3];
    const float* b1   = (const float*)d_in[4];
    const float* w2   = (const float*)d_in[5];
    const float* b2   = (const float*)d_in[6];
    const float* gam  = (const float*)d_in[7];
    const float* bet  = (const float*)d_in[8];

    float* out0 = (float*)d_out;                 // final `out` (also ping buffer S)
    float* out1 = out0 + (size_t)NNODES * DIM;   // final `r`   (also AGG buffer)
    float* norm = (float*)d_ws;                  // [NNODES] deg -> norm

    const int ND = NNODES * DIM;
    dim3 blk(256);
    dim3 gN((NNODES + 255) / 256);
    dim3 gE((NEDGES + 255) / 256);
    dim3 gND16((NNODES * 16 + 255) / 256);       // scale kernels (float4 per thread)
    dim3 gND((ND + 255) / 256);                  // zero big buffer
    dim3 gSc((NEDGES * 16 + 255) / 256);         // scatter (16 lanes per edge)

    // degree -> norm
    zero_kernel<<<gN, blk, 0, stream>>>(norm, NNODES);
    degree_kernel<<<gE, blk, 0, stream>>>(edst, norm, NEDGES);
    norm_kernel<<<gN, blk, 0, stream>>>(norm, NNODES);

    // hop 1
    scale_kernel<<<gND16, blk, 0, stream>>>(feat, norm, out0, NNODES, 1); // S = norm*h0
    zero_kernel<<<gND, blk, 0, stream>>>(out1, ND);
    scatter_kernel<<<gSc, blk, 0, stream>>>(out0, esrc, edst, out1, NEDGES);
    // hop 2 (S = norm^2 * AGG fuses h1 finalize + next-hop scaling)
    scale_kernel<<<gND16, blk, 0, stream>>>(out1, norm, out0, NNODES, 2);
    zero_kernel<<<gND, blk, 0, stream>>>(out1, ND);
    scatter_kernel<<<gSc, blk, 0, stream>>>(out0, esrc, edst, out1, NEDGES);
    // hop 3
    scale_kernel<<<gND16, blk, 0, stream>>>(out1, norm, out0, NNODES, 2);
    zero_kernel<<<gND, blk, 0, stream>>>(out1, ND);
    scatter_kernel<<<gSc, blk, 0, stream>>>(out0, esrc, edst, out1, NEDGES);
    // r = norm * AGG  (in place -> final r output)
    scale_kernel<<<gND16, blk, 0, stream>>>(out1, norm, out1, NNODES, 1);

    // LN + FFN + residuals -> out0
    ffn_wmma_kernel<<<dim3(NNODES / 128), blk, 0, stream>>>(
        out1, feat, w1, b1, w2, b2, gam, bet, out0);
}